// ConditionedPNA_13984413516119
// MI455X (gfx1250) — compile-verified
//
#include <hip/hip_runtime.h>
#include <math.h>

// ---------------------------------------------------------------------------
// ConditionedPNA for MI455X (gfx1250, wave32).
//  - Edge scatter: f32 atomics + ordered-uint atomics for max/min.
//  - Dense GEMMs: V_WMMA_F32_16X16X4_F32 (full f32 precision).
//    A layout (ISA 7.12.2): lane L<16 -> row L, K = k0,k0+1 ; lane L+16 ->
//    row L-16, K = k0+2,k0+3.  B assumed to mirror A's K interleave.
//    C/D: VGPR r -> row r (lanes 0-15) / row r+8 (lanes 16-31), col = lane%16.
//  - Weight staging: GLOBAL_LOAD_ASYNC_TO_LDS_B128 (direct HBM/L2 -> LDS DMA,
//    ASYNCcnt-tracked) instead of VGPR round-trips.
// ---------------------------------------------------------------------------

#define N_NODES 50000
#define N_EDGES 800000
#define DIM     64
#define N_REL   400
#define N_TAILS 512
#define NTILES  (N_NODES / 16)   // 3125, exact

typedef __attribute__((ext_vector_type(2))) float v2f;
typedef __attribute__((ext_vector_type(8))) float v8f;

static __device__ inline v8f wmma4(v2f a, v2f b, v8f c) {
  // 8 args: (neg_a, A, neg_b, B, c_mod, C, reuse_a, reuse_b)
  return __builtin_amdgcn_wmma_f32_16x16x4_f32(false, a, false, b, (short)0, c,
                                               false, false);
}

// Async global->LDS 16B copy (GVS mode): lds[vdst+off] = mem[saddr+vaddr+off]
static __device__ inline void async_copy_b128(unsigned lds_byte_addr,
                                              unsigned global_off,
                                              const float* sbase) {
  asm volatile("global_load_async_to_lds_b128 %0, %1, %2"
               :
               : "v"(lds_byte_addr), "v"(global_off), "s"(sbase)
               : "memory");
}
static __device__ inline void wait_async() {
  asm volatile("s_wait_asynccnt 0x0" ::: "memory");
}
// generic LDS pointer -> raw 32-bit LDS byte address
static __device__ inline unsigned lds_addr32(const void* p) {
  return (unsigned)(size_t)p;
}

// order-preserving float<->uint encoding for atomicMax/Min
static __device__ inline unsigned encf(float f) {
  unsigned u = __float_as_uint(f);
  return (u & 0x80000000u) ? ~u : (u ^ 0x80000000u);
}
static __device__ inline float decf(unsigned u) {
  u = (u & 0x80000000u) ? (u ^ 0x80000000u) : ~u;
  return __uint_as_float(u);
}
#define ENC_NEG_INF 0x007FFFFFu  // encf(-inf)
#define ENC_POS_INF 0xFF800000u  // encf(+inf)

// ---------------------------------------------------------------------------
__global__ void k_init(float* hidden, float* deg, float* scalars,
                       const float* head_embeds, const int* head_index) {
  int idx = blockIdx.x * blockDim.x + threadIdx.x;
  if (idx >= N_NODES * DIM) return;
  int node = idx >> 6, d = idx & 63;
  int h = *head_index;
  hidden[idx] = (node == h) ? head_embeds[d] : 0.0f;
  if (d == 0) deg[node] = 0.0f;
  if (idx == 0) scalars[0] = 0.0f;  // log-degree sum accumulator
}

__global__ void k_deg(const int* edge_dst, float* deg) {
  int e = blockIdx.x * blockDim.x + threadIdx.x;
  if (e >= N_EDGES) return;
  atomicAdd(&deg[edge_dst[e]], 1.0f);
}

__global__ void k_logsum(const float* deg, float* scalars) {
  __shared__ float red[256];
  int i = blockIdx.x * 256 + threadIdx.x;
  red[threadIdx.x] = (i < N_NODES) ? logf(deg[i] + 1.0f) : 0.0f;
  __syncthreads();
  for (int s = 128; s > 0; s >>= 1) {
    if (threadIdx.x < s) red[threadIdx.x] += red[threadIdx.x + s];
    __syncthreads();
  }
  if (threadIdx.x == 0) atomicAdd(&scalars[0], red[0]);
}

// score_fn for the zero vector and the head embedding (single block).
__global__ void k_score_init(const float* head_embeds, const float* rel_table,
                             const int* rI, const float* lin_W,
                             const float* lin_b, const float* W1,
                             const float* b1, const float* W2, const float* b2,
                             float* scalars) {
  __shared__ float q[DIM], x[DIM], redH[128], redZ[128];
  int t = threadIdx.x;  // 128 threads
  int r = *rI;
  if (t < DIM) q[t] = rel_table[r * DIM + t];
  __syncthreads();
  if (t < DIM) {
    float acc = lin_b[t];
    for (int k = 0; k < DIM; ++k) acc += head_embeds[k] * lin_W[k * DIM + t];
    for (int k = 0; k < DIM; ++k) acc += q[k] * lin_W[(DIM + k) * DIM + t];
    x[t] = head_embeds[t] * acc;  // hidden * heuristic
  }
  __syncthreads();
  float acc = b1[t];
  for (int k = 0; k < DIM; ++k) acc += x[k] * W1[k * 128 + t];
  redH[t] = fmaxf(acc, 0.0f) * W2[t];
  redZ[t] = fmaxf(b1[t], 0.0f) * W2[t];  // zero-vector case: h1 = relu(b1)
  __syncthreads();
  for (int s = 64; s > 0; s >>= 1) {
    if (t < s) { redH[t] += redH[t + s]; redZ[t] += redZ[t + s]; }
    __syncthreads();
  }
  if (t == 0) {
    float bb = b2[0];
    scalars[1] = fminf(fmaxf(redZ[0] + bb, -15.0f), 15.0f);  // zero_s
    scalars[2] = fminf(fmaxf(redH[0] + bb, -15.0f), 15.0f);  // head_s
  }
}

__global__ void k_scale_fill(const float* deg, const float* scalars,
                             const int* head_index, float* scale,
                             float* score) {
  int i = blockIdx.x * blockDim.x + threadIdx.x;
  if (i >= N_NODES) return;
  float mean = scalars[0] / (float)N_NODES;
  scale[i] = logf(deg[i] + 1.0f) / mean;
  score[i] = (i == *head_index) ? scalars[2] : scalars[1];
}

// per-layer: gate*hidden and re-init aggregate buffers
__global__ void k_prep(const float* hidden, const float* score, float* li,
                       float* sum, float* sumsq, unsigned* mx, unsigned* mn) {
  int idx = blockIdx.x * blockDim.x + threadIdx.x;
  if (idx >= N_NODES * DIM) return;
  int node = idx >> 6;
  float g = 1.0f / (1.0f + expf(-score[node]));
  li[idx] = g * hidden[idx];
  sum[idx] = 0.0f;
  sumsq[idx] = 0.0f;
  mx[idx] = ENC_NEG_INF;
  mn[idx] = ENC_POS_INF;
}

// one wave per edge; each lane owns 2 dims (float2)
__global__ void k_edge(const float* li, const float* rel_W, const int* esrc,
                       const int* edst, const int* etype, float* sum,
                       float* sumsq, unsigned* mx, unsigned* mn, int l) {
  int gid = blockIdx.x * blockDim.x + threadIdx.x;
  int e = gid >> 5, lane = gid & 31;
  if (e >= N_EDGES) return;
  int src = esrc[e], dst = edst[e], ty = etype[e];
  float2 a = *(const float2*)(li + (size_t)src * DIM + lane * 2);
  float2 w =
      *(const float2*)(rel_W + ((size_t)(l * N_REL + ty)) * DIM + lane * 2);
  float m0 = a.x * w.x, m1 = a.y * w.y;
  size_t o = (size_t)dst * DIM + lane * 2;
  atomicAdd(sum + o, m0);
  atomicAdd(sum + o + 1, m1);
  atomicAdd(sumsq + o, m0 * m0);
  atomicAdd(sumsq + o + 1, m1 * m1);
  atomicMax(mx + o, encf(m0));
  atomicMax(mx + o + 1, encf(m1));
  atomicMin(mn + o, encf(m0));
  atomicMin(mn + o + 1, encf(m1));
}

// update = feat[N,512] @ agg_W[512,64] + agg_b ; hidden += where(has_in,...)
// One wave per 16-node tile; feat built on the fly; agg_W staged 64KB x 2
// via async global->LDS DMA.
__global__ void k_update(float* hidden, const float* sum, const float* sumsq,
                         const unsigned* mx, const unsigned* mn,
                         const float* deg, const float* scale,
                         const float* agg_W, const float* agg_b, int l) {
  __shared__ float W_s[256 * DIM];  // 64KB: half the K rows of agg_W[l]
  int t = threadIdx.x;              // 128 threads = 4 waves
  const float* Wl = agg_W + (size_t)l * 512 * DIM;
  int wave = t >> 5, lane = t & 31;
  int tIdx = blockIdx.x * 4 + wave;
  bool active = tIdx < NTILES;
  int base = tIdx * 16, rL = lane & 15, half = lane >> 4;
  int rowNode = active ? (base + rL) : 0;

  float degv = deg[rowNode];
  float invd = 1.0f / fmaxf(degv, 1.0f);
  float sc = scale[rowNode];
  const float* rs = sum + (size_t)rowNode * DIM;
  const float* rq = sumsq + (size_t)rowNode * DIM;
  const unsigned* rmx = mx + (size_t)rowNode * DIM;
  const unsigned* rmn = mn + (size_t)rowNode * DIM;
  unsigned lbase = lds_addr32(W_s);

  v8f acc[4] = {};
  for (int ph = 0; ph < 2; ++ph) {
    __syncthreads();  // previous-phase readers done before overwrite
    for (int byte = t * 16; byte < 256 * DIM * 4; byte += 128 * 16)
      async_copy_b128(lbase + byte, (unsigned)(ph * 256 * DIM * 4 + byte), Wl);
    wait_async();
    __syncthreads();
    if (active) {
      for (int k0 = ph * 256; k0 < ph * 256 + 256; k0 += 4) {
        int blk = k0 >> 6, bb = blk & 3;       // uniform per k-step
        int d0 = (k0 & 63) + half * 2;         // dim within 64-block
        float vx, vy;
        if (bb == 0) {                         // mean
          float2 s = *(const float2*)(rs + d0);
          vx = s.x * invd; vy = s.y * invd;
        } else if (bb == 1) {                  // max
          uint2 u = *(const uint2*)(rmx + d0);
          vx = decf(u.x); vy = decf(u.y);
        } else if (bb == 2) {                  // min
          uint2 u = *(const uint2*)(rmn + d0);
          vx = decf(u.x); vy = decf(u.y);
        } else {                               // std
          float2 s = *(const float2*)(rs + d0);
          float2 q = *(const float2*)(rq + d0);
          float m0 = s.x * invd, m1 = s.y * invd;
          vx = __builtin_amdgcn_sqrtf(fmaxf(q.x * invd - m0 * m0, 0.0f) + 1e-6f);
          vy = __builtin_amdgcn_sqrtf(fmaxf(q.y * invd - m1 * m1, 0.0f) + 1e-6f);
        }
        if (blk >= 4) { vx *= sc; vy *= sc; }  // scaled copies (blocks 4-7)
        v2f a; a[0] = vx; a[1] = vy;
        int kb = (k0 - ph * 256) + half * 2;   // K row within W_s
#pragma unroll
        for (int tt = 0; tt < 4; ++tt) {
          int c = tt * 16 + rL;
          v2f b; b[0] = W_s[kb * DIM + c]; b[1] = W_s[(kb + 1) * DIM + c];
          acc[tt] = wmma4(a, b, acc[tt]);
        }
      }
    }
  }
  if (!active) return;
  const float* bl = agg_b + l * DIM;
#pragma unroll
  for (int tt = 0; tt < 4; ++tt) {
    int c = tt * 16 + rL;
    float bc = bl[c];
#pragma unroll
    for (int r = 0; r < 8; ++r) {
      int M = r + half * 8;
      float dM = __shfl(degv, M, 32);  // lane M holds deg[base+M]
      if (dM > 0.0f) {
        float* p = hidden + (size_t)(base + M) * DIM + c;
        *p += acc[tt][r] + bc;
      }
    }
  }
}

// score_fn over all nodes via two WMMA GEMMs per 16-node tile.
__global__ void k_score(const float* hidden, float* score, const float* deg,
                        const float* rel_table, const int* rI,
                        const float* lin_W, const float* lin_b,
                        const float* W1, const float* b1, const float* W2,
                        const float* b2) {
  __shared__ float wbuf[8192];           // 32KB: lin_W then W1
  __shared__ float xstage[4][16 * DIM];  // per-wave x tile, row-major 16x64
  __shared__ float b1_s[128], W2_s[128], linb_s[DIM], q_s[DIM];
  int t = threadIdx.x;  // 128 threads = 4 waves
  if (t < 128) { b1_s[t] = b1[t]; W2_s[t] = W2[t]; }
  if (t < DIM) { linb_s[t] = lin_b[t]; q_s[t] = rel_table[(*rI) * DIM + t]; }
  unsigned wb = lds_addr32(wbuf);
  for (int byte = t * 16; byte < 8192 * 4; byte += 128 * 16)
    async_copy_b128(wb + byte, (unsigned)byte, lin_W);  // [128][64]
  wait_async();
  __syncthreads();

  int wave = t >> 5, lane = t & 31;
  int tIdx = blockIdx.x * 4 + wave;
  bool active = tIdx < NTILES;
  int base = tIdx * 16, rL = lane & 15, half = lane >> 4;
  int rowNode = active ? (base + rL) : 0;
  float* xs = &xstage[wave][0];

  if (active) {
    // GEMM1: heuristic[16,64] = [hidden|q][16,128] @ lin_W[128,64]
    v8f acc[4] = {};
    for (int k0 = 0; k0 < 128; k0 += 4) {
      int kA = k0 + half * 2;
      v2f a;
      if (k0 < 64) {  // uniform branch (k0 multiple of 4)
        float2 h2 = *(const float2*)(hidden + (size_t)rowNode * DIM + kA);
        a[0] = h2.x; a[1] = h2.y;
      } else {
        a[0] = q_s[kA - 64]; a[1] = q_s[kA - 63];
      }
#pragma unroll
      for (int tt = 0; tt < 4; ++tt) {
        int c = tt * 16 + rL;
        v2f b; b[0] = wbuf[kA * DIM + c]; b[1] = wbuf[(kA + 1) * DIM + c];
        acc[tt] = wmma4(a, b, acc[tt]);
      }
    }
    // x = hidden * (heuristic + lin_b) -> LDS (per-wave tile)
#pragma unroll
    for (int tt = 0; tt < 4; ++tt) {
      int c = tt * 16 + rL;
      float lb = linb_s[c];
#pragma unroll
      for (int r = 0; r < 8; ++r) {
        int M = r + half * 8;
        float hv = hidden[(size_t)(base + M) * DIM + c];
        xs[M * DIM + c] = hv * (acc[tt][r] + lb);
      }
    }
  }
  __syncthreads();
  for (int byte = t * 16; byte < 8192 * 4; byte += 128 * 16)
    async_copy_b128(wb + byte, (unsigned)byte, W1);  // [64][128]
  wait_async();
  __syncthreads();
  if (!active) return;

  // GEMM2: h1[16,128] = x[16,64] @ W1[64,128]
  v8f acc2[8] = {};
  for (int k0 = 0; k0 < 64; k0 += 4) {
    int kA = k0 + half * 2;
    v2f a; a[0] = xs[rL * DIM + kA]; a[1] = xs[rL * DIM + kA + 1];
#pragma unroll
    for (int tt = 0; tt < 8; ++tt) {
      int c = tt * 16 + rL;
      v2f b; b[0] = wbuf[kA * 128 + c]; b[1] = wbuf[(kA + 1) * 128 + c];
      acc2[tt] = wmma4(a, b, acc2[tt]);
    }
  }
  // s = relu(h1 + b1) . W2 + b2 ; reduce 16 lanes per half-wave
  float part[8];
#pragma unroll
  for (int r = 0; r < 8; ++r) part[r] = 0.0f;
#pragma unroll
  for (int tt = 0; tt < 8; ++tt) {
    int c = tt * 16 + rL;
    float w2 = W2_s[c], bb = b1_s[c];
#pragma unroll
    for (int r = 0; r < 8; ++r)
      part[r] += fmaxf(acc2[tt][r] + bb, 0.0f) * w2;
  }
  for (int off = 1; off < 16; off <<= 1)
#pragma unroll
    for (int r = 0; r < 8; ++r) part[r] += __shfl_xor(part[r], off, 32);
  if (rL == 0) {
    float b2v = b2[0];
#pragma unroll
    for (int r = 0; r < 8; ++r) {
      int node = base + half * 8 + r;
      if (deg[node] > 0.0f) score[node] = part[r] + b2v;  // else keep old
    }
  }
}

__global__ void k_out(const float* score, const int* t_index, float* out) {
  int j = blockIdx.x * blockDim.x + threadIdx.x;
  if (j >= N_TAILS) return;
  out[j] = score[t_index[j]];
}

// ---------------------------------------------------------------------------
extern "C" void kernel_launch(void* const* d_in, const int* in_sizes, int n_in,
                              void* d_out, int out_size, void* d_ws,
                              size_t ws_size, hipStream_t stream) {
  const float* head_embeds = (const float*)d_in[0];
  const float* rel_table   = (const float*)d_in[1];
  const float* rel_W       = (const float*)d_in[2];
  const float* agg_W       = (const float*)d_in[3];
  const float* agg_b       = (const float*)d_in[4];
  const float* lin_W       = (const float*)d_in[5];
  const float* lin_b       = (const float*)d_in[6];
  const float* mlp_W1      = (const float*)d_in[7];
  const float* mlp_b1      = (const float*)d_in[8];
  const float* mlp_W2      = (const float*)d_in[9];
  const float* mlp_b2      = (const float*)d_in[10];
  const int* edge_src      = (const int*)d_in[11];
  const int* edge_dst      = (const int*)d_in[12];
  const int* edge_type     = (const int*)d_in[13];
  const int* head_index    = (const int*)d_in[14];
  const int* r_index       = (const int*)d_in[15];
  const int* t_index       = (const int*)d_in[16];
  float* out = (float*)d_out;

  char* ws = (char*)d_ws;
  size_t off = 0;
  auto take = [&](size_t bytes) {
    void* p = ws + off;
    off = (off + bytes + 255) & ~(size_t)255;
    return p;
  };
  const size_t ND = (size_t)N_NODES * DIM * sizeof(float);
  float* hidden   = (float*)take(ND);
  float* li       = (float*)take(ND);
  float* sum      = (float*)take(ND);
  float* sumsq    = (float*)take(ND);
  unsigned* mxb   = (unsigned*)take(ND);
  unsigned* mnb   = (unsigned*)take(ND);
  float* deg      = (float*)take(N_NODES * sizeof(float));
  float* scale    = (float*)take(N_NODES * sizeof(float));
  float* score    = (float*)take(N_NODES * sizeof(float));
  float* scalars  = (float*)take(16 * sizeof(float));

  const int TPB = 256;
  int gND = (N_NODES * DIM + TPB - 1) / TPB;
  int gN = (N_NODES + TPB - 1) / TPB;
  int gE = (N_EDGES + TPB - 1) / TPB;
  int gE32 = (N_EDGES * 32 + TPB - 1) / TPB;
  int gTile = (NTILES + 3) / 4;

  k_init<<<gND, TPB, 0, stream>>>(hidden, deg, scalars, head_embeds,
                                  head_index);
  k_score_init<<<1, 128, 0, stream>>>(head_embeds, rel_table, r_index, lin_W,
                                      lin_b, mlp_W1, mlp_b1, mlp_W2, mlp_b2,
                                      scalars);
  k_deg<<<gE, TPB, 0, stream>>>(edge_dst, deg);
  k_logsum<<<gN, TPB, 0, stream>>>(deg, scalars);
  k_scale_fill<<<gN, TPB, 0, stream>>>(deg, scalars, head_index, scale, score);

  for (int l = 0; l < 3; ++l) {
    k_prep<<<gND, TPB, 0, stream>>>(hidden, score, li, sum, sumsq, mxb, mnb);
    k_edge<<<gE32, TPB, 0, stream>>>(li, rel_W, edge_src, edge_dst, edge_type,
                                     sum, sumsq, mxb, mnb, l);
    k_update<<<gTile, 128, 0, stream>>>(hidden, sum, sumsq, mxb, mnb, deg,
                                        scale, agg_W, agg_b, l);
    k_score<<<gTile, 128, 0, stream>>>(hidden, score, deg, rel_table, r_index,
                                       lin_W, lin_b, mlp_W1, mlp_b1, mlp_W2,
                                       mlp_b2);
  }
  k_out<<<(N_TAILS + TPB - 1) / TPB, TPB, 0, stream>>>(score, t_index, out);
}